// MyModel_87454124082148
// MI455X (gfx1250) — compile-verified
//
#include <hip/hip_runtime.h>

typedef __attribute__((ext_vector_type(2))) float v2f;
typedef __attribute__((ext_vector_type(8))) float v8f;

static constexpr int ROW_LEN     = 4096;
static constexpr int T           = 4;            // 16x16 tiles per wave block
static constexpr int COLW        = 16 * T;       // block row width = 64
static constexpr int BLOCK_ELEMS = 16 * COLW;    // 1024 elements per wave
static constexpr int WAVES       = ROW_LEN / BLOCK_ELEMS;   // 4
static constexpr int THREADS     = WAVES * 32;   // 128 (wave32)

// Identity cross-swap of the two 16-lane halves of a wave32 (pure VALU).
__device__ __forceinline__ float swap_half(float v) {
    unsigned u = __float_as_uint(v);
    unsigned r = __builtin_amdgcn_permlanex16(u, u, 0x76543210u, 0xFEDCBA98u,
                                              false, false);
    return __uint_as_float(r);
}

// Inclusive cumsum along a 4096-float row; one workgroup (4 waves) per row.
// Each wave owns a 16x64 block (flat idx = 64*i + 16*t + j):
//   acc_t = X_t @ U + P        (U upper-tri ones: within-row prefix,
//                               P = (sum_{s<t} X_s) @ J: earlier-tile offsets)
//   O     = Lstrict @ Q        (Q[r][*] = rowTotal(r): across-block-row offsets)
//   + LDS exchange of the 4 wave totals (only DS traffic in the kernel).
__global__ __launch_bounds__(THREADS)
void cumsum_row_wmma(const float* __restrict__ x, float* __restrict__ y, int nrows)
{
    __shared__ float waveTot[WAVES];

    const int row = blockIdx.x;
    if (row >= nrows) return;                     // uniform across the block

    const float* __restrict__ xrow = x + (size_t)row * ROW_LEN;
    float*       __restrict__ yrow = y + (size_t)row * ROW_LEN;

    const int tid  = threadIdx.x;
    const int wave = tid >> 5;
    const int lane = tid & 31;
    const int half = lane >> 4;                   // 0: lanes 0-15, 1: lanes 16-31
    const int lrow = lane & 15;                   // A row / B,D column index

    // Constant operands. A(16x4) layout: lane holds K = 4k+2*half+{0,1} of row lrow.
    // B(4x16)  layout: lane holds rows 4k+2*half+{0,1} of column lrow.
    v2f bU[4], aL[4];
    const v2f bJ = {1.0f, 1.0f};
#pragma unroll
    for (int k = 0; k < 4; ++k) {
        const int r0 = 4 * k + 2 * half;
        bU[k].x = (r0     <= lrow) ? 1.0f : 0.0f; // U[r][c] = (r <= c)
        bU[k].y = (r0 + 1 <= lrow) ? 1.0f : 0.0f;
        aL[k].x = (r0     <  lrow) ? 1.0f : 0.0f; // Lstrict[row=lrow][col=r0..]
        aL[k].y = (r0 + 1 <  lrow) ? 1.0f : 0.0f;
    }

    // Per-lane A-operand base: X_t[lrow][4k+2*half] = block[COLW*lrow + 16t + 4k + 2*half]
    const float* abase = xrow + wave * BLOCK_ELEMS + COLW * lrow + 2 * half;

    v8f acc[T];
    v8f P = {0.f, 0.f, 0.f, 0.f, 0.f, 0.f, 0.f, 0.f};

#pragma unroll
    for (int t = 0; t < T; ++t) {
        v2f a[4];
#pragma unroll
        for (int k = 0; k < 4; ++k)
            a[k] = *(const v2f*)(abase + 16 * t + 4 * k);

        // acc_t = X_t @ U + (earlier tiles' row sums)
        v8f s = P;
#pragma unroll
        for (int k = 0; k < 4; ++k)
            s = __builtin_amdgcn_wmma_f32_16x16x4_f32(
                    false, a[k], false, bU[k], (short)0, s, false, false);
        acc[t] = s;

        // P += X_t @ J  (running row totals, D layout)
#pragma unroll
        for (int k = 0; k < 4; ++k)
            P = __builtin_amdgcn_wmma_f32_16x16x4_f32(
                    false, a[k], false, bJ, (short)0, P, false, false);
    }

    // P: lane n, reg v = rowTotal(v + 8*half) over the full 64-wide rows.
    float own[8], oth[8];
#pragma unroll
    for (int v = 0; v < 8; ++v) {
        own[v] = P[v];
        oth[v] = swap_half(P[v]);                 // other half's rows
    }

    // Q (B operand): Q[k] = { rowTot(4k+2*half), rowTot(4k+2*half+1) }
    // rowTot(i) = (i>>3 == half) ? own[i&7] : oth[i&7]
    v2f Q[4];
#pragma unroll
    for (int k = 0; k < 4; ++k) {
        if (k < 2) {
            Q[k].x = half ? oth[4 * k + 2] : own[4 * k];
            Q[k].y = half ? oth[4 * k + 3] : own[4 * k + 1];
        } else {
            Q[k].x = half ? own[4 * k - 6] : oth[4 * k - 8];
            Q[k].y = half ? own[4 * k - 5] : oth[4 * k - 7];
        }
    }

    // O = Lstrict @ Q : exclusive prefix across the 16 block rows, D layout.
    v8f O = {0.f, 0.f, 0.f, 0.f, 0.f, 0.f, 0.f, 0.f};
#pragma unroll
    for (int k = 0; k < 4; ++k)
        O = __builtin_amdgcn_wmma_f32_16x16x4_f32(
                false, aL[k], false, Q[k], (short)0, O, false, false);

    // Block total = sum of all 16 row totals (identical in every lane).
    float tot = 0.f;
#pragma unroll
    for (int v = 0; v < 8; ++v) tot += own[v] + oth[v];

    if (lane == 0) waveTot[wave] = tot;
    __syncthreads();

    float off = 0.f;
#pragma unroll
    for (int w = 0; w < WAVES; ++w)
        if (w < wave) off += waveTot[w];

    v8f OO;
#pragma unroll
    for (int v = 0; v < 8; ++v) OO[v] = O[v] + off;

    // Store: D layout -> element (i = v + 8*half, col = 16t + lrow).
    float* obase = yrow + wave * BLOCK_ELEMS + COLW * 8 * half + lrow;
#pragma unroll
    for (int t = 0; t < T; ++t)
#pragma unroll
        for (int v = 0; v < 8; ++v)
            obase[COLW * v + 16 * t] = acc[t][v] + OO[v];
}

extern "C" void kernel_launch(void* const* d_in, const int* in_sizes, int n_in,
                              void* d_out, int out_size, void* d_ws, size_t ws_size,
                              hipStream_t stream) {
    (void)n_in; (void)out_size; (void)d_ws; (void)ws_size;
    const float* x = (const float*)d_in[0];
    float*       y = (float*)d_out;
    const int nrows = in_sizes[0] / ROW_LEN;      // 8*4096 = 32768 rows
    hipLaunchKernelGGL(cumsum_row_wmma, dim3(nrows), dim3(THREADS), 0, stream,
                       x, y, nrows);
}